// ARDecoderCell_68393059221786
// MI455X (gfx1250) — compile-verified
//
#include <hip/hip_runtime.h>
#include <hip/hip_bf16.h>
#include <math.h>

// Problem dims
#define B_   32
#define T_   1024
#define ENC_ 1024
#define DEC_ 1024
#define MEL_ 80
#define KIN_ (MEL_ + ENC_)        // 1104
#define KTOT (KIN_ + DEC_)        // 2128
#define NZ   (4 * DEC_)           // 4096

typedef __attribute__((ext_vector_type(2))) float v2f;
typedef __attribute__((ext_vector_type(8))) float v8f;

// ---------------------------------------------------------------------------
// Kernel 1: additive-attention scores[b,t] = sum_d scale[d]*tanh(h[b,d]+enc[b,t,d])
// grid (B, T/64), block 256 (8 waves, 8 t's per wave)
// ---------------------------------------------------------------------------
__global__ __launch_bounds__(256)
void k_scores(const float* __restrict__ enc, const float* __restrict__ h,
              const float* __restrict__ scale, float* __restrict__ scores) {
    __shared__ float h_l[ENC_];
    __shared__ float s_l[ENC_];
    const int b  = blockIdx.x;
    const int t0 = blockIdx.y * 64;
    for (int i = threadIdx.x; i < ENC_; i += 256) {
        h_l[i] = h[b * DEC_ + i];
        s_l[i] = scale[i];
    }
    __syncthreads();
    const int wave = threadIdx.x >> 5;
    const int lane = threadIdx.x & 31;
    for (int it = 0; it < 8; ++it) {
        const int t = t0 + wave * 8 + it;
        const float* er = enc + ((size_t)b * T_ + t) * ENC_;
        float s = 0.f;
#pragma unroll 4
        for (int d = lane; d < ENC_; d += 32)
            s += s_l[d] * tanhf(h_l[d] + er[d]);
        for (int off = 16; off; off >>= 1) s += __shfl_xor(s, off, 32);
        if (lane == 0) scores[b * T_ + t] = s;
    }
}

// ---------------------------------------------------------------------------
// Kernel 2: in-place softmax over T for each batch row.  grid B, block 256.
// ---------------------------------------------------------------------------
__global__ __launch_bounds__(256)
void k_softmax(float* __restrict__ sc) {
    __shared__ float red_max[8];
    __shared__ float red_sum[8];
    const int b   = blockIdx.x;
    const int tid = threadIdx.x;
    const int lane = tid & 31, wave = tid >> 5;

    float v[4];
    float m = -INFINITY;
#pragma unroll
    for (int i = 0; i < 4; ++i) {
        v[i] = sc[b * T_ + tid + i * 256];
        m = fmaxf(m, v[i]);
    }
    for (int off = 16; off; off >>= 1) m = fmaxf(m, __shfl_xor(m, off, 32));
    if (lane == 0) red_max[wave] = m;
    __syncthreads();
    m = red_max[0];
#pragma unroll
    for (int w = 1; w < 8; ++w) m = fmaxf(m, red_max[w]);

    float s = 0.f;
#pragma unroll
    for (int i = 0; i < 4; ++i) {
        v[i] = __expf(v[i] - m);
        s += v[i];
    }
    for (int off = 16; off; off >>= 1) s += __shfl_xor(s, off, 32);
    if (lane == 0) red_sum[wave] = s;
    __syncthreads();
    s = red_sum[0];
#pragma unroll
    for (int w = 1; w < 8; ++w) s += red_sum[w];
    const float inv = 1.f / s;
#pragma unroll
    for (int i = 0; i < 4; ++i)
        sc[b * T_ + tid + i * 256] = v[i] * inv;
}

// ---------------------------------------------------------------------------
// Kernel 3: context[b,d] = sum_t attn[b,t]*enc[b,t,d].  grid (B, ENC/256), block 256.
// enc second pass: expected L2-resident (128MB of 192MB L2).
// ---------------------------------------------------------------------------
__global__ __launch_bounds__(256)
void k_context(const float* __restrict__ enc, const float* __restrict__ attn,
               float* __restrict__ ctx) {
    __shared__ float a_l[T_];
    const int b  = blockIdx.x;
    const int d0 = blockIdx.y * 256;
    for (int i = threadIdx.x; i < T_; i += 256) a_l[i] = attn[b * T_ + i];
    __syncthreads();
    const float* eb = enc + (size_t)b * T_ * ENC_ + d0 + threadIdx.x;
    float acc = 0.f;
#pragma unroll 8
    for (int t = 0; t < T_; ++t)
        acc += a_l[t] * eb[(size_t)t * ENC_];
    ctx[b * ENC_ + d0 + threadIdx.x] = acc;
}

// ---------------------------------------------------------------------------
// Kernel 4: z[32,4096] = [prev_mel | context | h] @ [kernel ; rec_kernel]
// FP32 WMMA 16x16x4.  grid NZ/64 blocks, 256 threads (8 waves = 2 Mtiles x 4 Ntiles).
// K staged in LDS panels of 64; strides chosen for conflict-free fragment reads.
// ---------------------------------------------------------------------------
#define KP 64
#define XS 40   // xT row stride (2*XS % 64 == 16 -> banks disjoint between half-waves)
#define WS 72   // W  row stride (2*WS % 64 == 16)

__global__ __launch_bounds__(256)
void k_zgemm(const float* __restrict__ pm, const float* __restrict__ ctx,
             const float* __restrict__ h,  const float* __restrict__ Wk,
             const float* __restrict__ Wr, float* __restrict__ z) {
    __shared__ float xT[KP * XS];
    __shared__ float Wl[KP * WS];
    const int n0   = blockIdx.x * 64;
    const int tid  = threadIdx.x;
    const int wave = tid >> 5, lane = tid & 31;
    const int mt = wave >> 2, nt = wave & 3;
    const int l15 = lane & 15;
    const int hi  = lane >> 4;          // 0: lanes 0-15, 1: lanes 16-31

    v8f acc = {};

    for (int K0 = 0; K0 < 2176; K0 += KP) {   // 34 panels covers KTOT=2128 (zero-padded)
        // stage x-transposed panel: xT[k][m] = x_combined[m][K0+k]
        for (int i = tid; i < KP * 32; i += 256) {
            const int k = i >> 5, m = i & 31;
            const int K = K0 + k;
            float val = 0.f;
            if (K < MEL_)       val = pm[m * MEL_ + K];
            else if (K < KIN_)  val = ctx[m * ENC_ + (K - MEL_)];
            else if (K < KTOT)  val = h[m * DEC_ + (K - KIN_)];
            xT[k * XS + m] = val;
        }
        // stage weight panel Wl[k][0..63] = W[K0+k][n0..n0+63]
        for (int i = tid; i < KP * 16; i += 256) {
            const int k = i >> 4, c4 = (i & 15) * 4;
            const int K = K0 + k;
            float4 val = make_float4(0.f, 0.f, 0.f, 0.f);
            if (K < KIN_)
                val = *(const float4*)(Wk + (size_t)K * NZ + n0 + c4);
            else if (K < KTOT)
                val = *(const float4*)(Wr + (size_t)(K - KIN_) * NZ + n0 + c4);
            *(float4*)&Wl[k * WS + c4] = val;
        }
        __syncthreads();
#pragma unroll
        for (int ks = 0; ks < KP; ks += 4) {
            // A 16x4 frag: lanes0-15 hold K=ks,ks+1; lanes16-31 hold K=ks+2,ks+3
            // B 4x16 frag: same K split, N striped across the 16 lanes of each half
            const int ar = ks + 2 * hi;
            v2f a, bf;
            a.x  = xT[ar * XS       + mt * 16 + l15];
            a.y  = xT[(ar + 1) * XS + mt * 16 + l15];
            bf.x = Wl[ar * WS       + nt * 16 + l15];
            bf.y = Wl[(ar + 1) * WS + nt * 16 + l15];
            acc = __builtin_amdgcn_wmma_f32_16x16x4_f32(
                false, a, false, bf, (short)0, acc, false, false);
        }
        __syncthreads();
    }
    // C/D layout: lanes0-15: N=lane, M=r ; lanes16-31: N=lane-16, M=8+r
    const int n     = n0 + nt * 16 + l15;
    const int mbase = mt * 16 + hi * 8;
#pragma unroll
    for (int r = 0; r < 8; ++r)
        z[(mbase + r) * NZ + n] = acc[r];
}

// ---------------------------------------------------------------------------
// Kernel 5: LSTM pointwise.  grid B*DEC/256 blocks.
// ---------------------------------------------------------------------------
__global__ __launch_bounds__(256)
void k_lstm_pw(const float* __restrict__ z, const float* __restrict__ bias,
               const float* __restrict__ c_old, float* __restrict__ h_new,
               float* __restrict__ c_new) {
    const int idx = blockIdx.x * 256 + threadIdx.x;    // 0..32767
    const int b = idx >> 10, d = idx & 1023;
    const float* zb = z + b * NZ;
    const float zi = zb[d]            + bias[d];
    const float zf = zb[DEC_ + d]     + bias[DEC_ + d];
    const float zg = zb[2 * DEC_ + d] + bias[2 * DEC_ + d];
    const float zo = zb[3 * DEC_ + d] + bias[3 * DEC_ + d];
    const float ig = 1.f / (1.f + __expf(-zi));
    const float fg = 1.f / (1.f + __expf(-zf));
    const float og = 1.f / (1.f + __expf(-zo));
    const float cn = fg * c_old[idx] + ig * tanhf(zg);
    const float hn = og * tanhf(cn);
    c_new[idx] = cn;
    h_new[idx] = hn;
}

// ---------------------------------------------------------------------------
// Kernel 6: mel[b,m] = h_new[b,:] @ proj_w[:,m] + proj_b[m].  grid B, block 128.
// ---------------------------------------------------------------------------
__global__ __launch_bounds__(128)
void k_proj(const float* __restrict__ h_new, const float* __restrict__ pw,
            const float* __restrict__ pb, float* __restrict__ mel) {
    __shared__ float h_l[DEC_];
    const int b = blockIdx.x;
    for (int i = threadIdx.x; i < DEC_; i += 128) h_l[i] = h_new[b * DEC_ + i];
    __syncthreads();
    const int m = threadIdx.x;
    if (m < MEL_) {
        float acc = pb[m];
#pragma unroll 8
        for (int d = 0; d < DEC_; ++d)
            acc += h_l[d] * pw[d * MEL_ + m];
        mel[b * MEL_ + m] = acc;
    }
}

// ---------------------------------------------------------------------------
extern "C" void kernel_launch(void* const* d_in, const int* in_sizes, int n_in,
                              void* d_out, int out_size, void* d_ws, size_t ws_size,
                              hipStream_t stream) {
    const float* prev_mel = (const float*)d_in[0];   // [32,80]
    const float* enc      = (const float*)d_in[1];   // [32,1024,1024]
    const float* h        = (const float*)d_in[2];   // [32,1024]
    const float* c        = (const float*)d_in[3];   // [32,1024]
    const float* scale    = (const float*)d_in[4];   // [1024]
    const float* Wk       = (const float*)d_in[5];   // [1104,4096]
    const float* Wr       = (const float*)d_in[6];   // [1024,4096]
    const float* bias     = (const float*)d_in[7];   // [4096]
    const float* pw       = (const float*)d_in[8];   // [1024,80]
    const float* pb       = (const float*)d_in[9];   // [80]

    float* ws   = (float*)d_ws;
    float* attn = ws;                       // B*T      (scores -> softmax in place)
    float* ctx  = ws + B_ * T_;             // B*ENC
    float* z    = ws + 2 * B_ * T_;         // B*NZ

    float* out   = (float*)d_out;
    float* mel   = out;                     // B*MEL
    float* h_new = out + B_ * MEL_;         // B*DEC
    float* c_new = out + B_ * MEL_ + B_ * DEC_;

    k_scores <<<dim3(B_, T_ / 64), 256, 0, stream>>>(enc, h, scale, attn);
    k_softmax<<<B_, 256, 0, stream>>>(attn);
    k_context<<<dim3(B_, ENC_ / 256), 256, 0, stream>>>(enc, attn, ctx);
    k_zgemm  <<<NZ / 64, 256, 0, stream>>>(prev_mel, ctx, h, Wk, Wr, z);
    k_lstm_pw<<<(B_ * DEC_) / 256, 256, 0, stream>>>(z, bias, c, h_new, c_new);
    k_proj   <<<B_, 128, 0, stream>>>(h_new, pw, pb, mel);
}